// YOLOLoss_40183714021890
// MI455X (gfx1250) — compile-verified
//
#include <hip/hip_runtime.h>
#include <stdint.h>

// ---------------- problem constants (fixed by the reference) ----------------
#define TPB              256                 // 8 wave32 waves per workgroup
#define CELLS_PER_CHUNK  256                 // one cell per thread per chunk
#define FLOATS_PER_CELL  30                  // D = B*5 + C = 2*5 + 20
#define BYTES_PER_CELL   120
#define CHUNK_TENSOR_BYTES (CELLS_PER_CHUNK * BYTES_PER_CELL)   // 30720 B per tensor
#define CHUNK_TOTAL_BYTES  (2 * CHUNK_TENSOR_BYTES)             // 61440 B (preds+targets)
#define LOADS_PER_THREAD   15                // 15 x b128 per thread per chunk
#define NUM_CLASSES      20
#define LAMBDA_NOOBJ     0.5f

// ---------------- CDNA5 async wait --------------------------------------
#if __has_builtin(__builtin_amdgcn_s_wait_asynccnt)
#define WAIT_ASYNCCNT(n) __builtin_amdgcn_s_wait_asynccnt(n)
#else
#define WAIT_ASYNCCNT(n) asm volatile("s_wait_asynccnt %0" :: "n"(n) : "memory")
#endif

// One chunk's async global->LDS DMA. Wave-uniform split: tid<128 stream preds,
// tid>=128 stream targets (128 thr x 15 x 16B = 30720B per tensor, exact).
// SADDR form: scalar 64-bit base, one 32-bit VGPR offset, and the 24-bit
// instruction offset (applied to BOTH global and LDS addresses per the ISA
// async pseudocode) varies across the 15 issues -> minimal VALU per load.
__device__ __forceinline__ void issue_chunk_loads(uint64_t srcBase,  // wave-uniform
                                                  uint32_t voff,     // chunkByteBase + k*16
                                                  uint32_t vdst) {   // ldsBuf + region + k*16
#pragma unroll
  for (int i = 0; i < LOADS_PER_THREAD; ++i) {
    asm volatile("global_load_async_to_lds_b128 %0, %1, %2 offset:%3"
                 :: "v"(vdst), "v"(voff), "s"(srcBase), "n"(i * 128 * 16)
                 : "memory");
  }
}

// Per-cell YOLO loss. For the LDS path: word stride 30 over 64 banks -> 32
// distinct banks per wave32, conflict-free.
__device__ __forceinline__ float cell_loss(const float* __restrict__ P,
                                           const float* __restrict__ T) {
  float iou[2];
#pragma unroll
  for (int j = 0; j < 2; ++j) {
    const float x1 = P[j * 5 + 0], y1 = P[j * 5 + 1], w1 = P[j * 5 + 2], h1 = P[j * 5 + 3];
    const float x2 = T[j * 5 + 0], y2 = T[j * 5 + 1], w2 = T[j * 5 + 2], h2 = T[j * 5 + 3];
    const float tlx = fmaxf(x1 - w1 * 0.5f, x2 - w2 * 0.5f);
    const float tly = fmaxf(y1 - h1 * 0.5f, y2 - h2 * 0.5f);
    const float brx = fminf(x1 + w1 * 0.5f, x2 + w2 * 0.5f);
    const float bry = fminf(y1 + h1 * 0.5f, y2 + h2 * 0.5f);
    const float iw = fmaxf(0.0f, brx - tlx);
    const float ih = fmaxf(0.0f, bry - tly);
    const float inter = iw * ih;
    const float uni = w1 * h1 + w2 * h2 - inter;
    iou[j] = inter / uni;
  }
  const int bidx = (iou[1] > iou[0]) ? 1 : 0;          // jnp.argmax: first max wins

  float class_loss = 0.0f;
  bool  hasObj = false;
  float bestT = T[10];
  int   gnd = 0;
#pragma unroll
  for (int c = 0; c < NUM_CLASSES; ++c) {
    const float pc = P[10 + c];
    const float tc = T[10 + c];
    const float d = pc - tc;
    class_loss += d * d;
    hasObj = hasObj || (tc != 0.0f);
    if (tc > bestT) { bestT = tc; gnd = c; }           // first-max semantics
  }
  const float pg = P[10 + gnd];

  const float dx = P[bidx * 5 + 0] - T[bidx * 5 + 0];
  const float dy = P[bidx * 5 + 1] - T[bidx * 5 + 1];
  const float pos_loss = hasObj ? (dx * dx + dy * dy) : 0.0f;

  float conf_loss = 0.0f;
#pragma unroll
  for (int j = 0; j < 2; ++j) {
    const float cd = iou[j] * pg - iou[j];
    const float w  = (j == bidx) ? 1.0f : LAMBDA_NOOBJ;
    conf_loss += w * cd * cd;
  }
  return pos_loss + class_loss + conf_loss;
}

__global__ void __launch_bounds__(TPB)
yolo_loss_partial(const float* __restrict__ preds,
                  const float* __restrict__ targets,
                  long long nCells, int nChunks,
                  float* __restrict__ partials) {
  // Double-buffered staging: 2 x 61440 B = 120 KB (2 workgroups / 320 KB WGP LDS)
  __shared__ float lds[2][CHUNK_TOTAL_BYTES / 4];
  __shared__ float red[TPB];

  const int tid = threadIdx.x;
  const int k = tid & 127;
  const bool isPred = (tid < 128);                      // wave-uniform (waves 0-3 vs 4-7)
  const uint64_t srcBase = (uint64_t)(uintptr_t)(isPred ? preds : targets);
  const uint32_t kOff = (uint32_t)k * 16u;
  const uint32_t region = isPred ? 0u : (uint32_t)CHUNK_TENSOR_BYTES;
  const uint32_t dstBase0 = (uint32_t)(uintptr_t)&lds[0][0] + region + kOff;
  const uint32_t dstBase1 = (uint32_t)(uintptr_t)&lds[1][0] + region + kOff;

  float acc = 0.0f;
  int pb = 0;
  const int c0 = (int)blockIdx.x;

  // Prologue: start DMA for this block's first chunk.
  if (c0 < nChunks)
    issue_chunk_loads(srcBase, (uint32_t)c0 * CHUNK_TENSOR_BYTES + kOff, dstBase0);

  for (int c = c0; c < nChunks; c += (int)gridDim.x) {
    const int nc = c + (int)gridDim.x;
    if (nc < nChunks) {
      // Kick next chunk's DMA into the other buffer (WAR safe: the trailing
      // barrier of the previous iteration covered all readers), then wait
      // until only those 15 new loads remain outstanding.
      issue_chunk_loads(srcBase, (uint32_t)nc * CHUNK_TENSOR_BYTES + kOff,
                        pb ? dstBase0 : dstBase1);
      WAIT_ASYNCCNT(LOADS_PER_THREAD);
    } else {
      WAIT_ASYNCCNT(0);
    }
    __syncthreads();   // all waves' DMA for chunk c has landed in LDS

    const float* P = &lds[pb][tid * FLOATS_PER_CELL];
    const float* T = &lds[pb][CHUNK_TENSOR_BYTES / 4 + tid * FLOATS_PER_CELL];
    acc += cell_loss(P, T);

    __syncthreads();   // safe to overwrite buffer pb next iteration
    pb ^= 1;
  }

  // Tail cells (nCells % 256, absent for this harness) -> block 0, direct global.
  if (blockIdx.x == 0) {
    const long long cell = (long long)nChunks * CELLS_PER_CHUNK + tid;
    if (cell < nCells)
      acc += cell_loss(preds + cell * FLOATS_PER_CELL,
                       targets + cell * FLOATS_PER_CELL);
  }

  // Deterministic fixed-order block tree reduction.
  red[tid] = acc;
  __syncthreads();
  for (int t = TPB / 2; t > 0; t >>= 1) {
    if (tid < t) red[tid] += red[tid + t];
    __syncthreads();
  }
  if (tid == 0) partials[blockIdx.x] = red[0];
}

__global__ void __launch_bounds__(TPB)
reduce_partials(const float* __restrict__ partials, int n, float invN,
                float* __restrict__ out) {
  __shared__ float red[TPB];
  float s = 0.0f;
  for (int i = threadIdx.x; i < n; i += TPB) s += partials[i];   // fixed order
  red[threadIdx.x] = s;
  __syncthreads();
  for (int t = TPB / 2; t > 0; t >>= 1) {
    if (threadIdx.x < t) red[threadIdx.x] += red[threadIdx.x + t];
    __syncthreads();
  }
  if (threadIdx.x == 0) out[0] = red[0] * invN;
}

extern "C" void kernel_launch(void* const* d_in, const int* in_sizes, int n_in,
                              void* d_out, int out_size, void* d_ws, size_t ws_size,
                              hipStream_t stream) {
  (void)n_in; (void)out_size; (void)ws_size;
  const float* preds   = (const float*)d_in[0];
  const float* targets = (const float*)d_in[1];

  const long long total  = (long long)in_sizes[0];                 // N*S*S*D
  const long long nCells = total / FLOATS_PER_CELL;                // N*S*S = 802816
  const int nChunks = (int)(nCells / CELLS_PER_CHUNK);             // full chunks (3136)
  const float invN  = 1.0f / (float)(nCells / 49);                 // /N (S=7)

  int grid = 392;                 // 3136 chunks -> 8 chunks/block pipeline depth
  if (grid > nChunks) grid = nChunks;
  if (grid < 1) grid = 1;         // tail-only safety

  float* partials = (float*)d_ws;
  yolo_loss_partial<<<grid, TPB, 0, stream>>>(preds, targets, nCells, nChunks, partials);
  reduce_partials<<<1, TPB, 0, stream>>>(partials, grid, invN, (float*)d_out);
}